// ReprogrammingLayer2_43112881717727
// MI455X (gfx1250) — compile-verified
//
#include <hip/hip_runtime.h>
#include <stdint.h>
#include <math.h>

// ---------------------------------------------------------------------------
// ReprogrammingLayer (cross-attention + output projection), fp32, CDNA5.
// - GEMMs via V_WMMA_F32_16X16X4_F32 (exact fp32 numerics)
// - All LDS tile staging via the Tensor Data Mover (TENSOR_LOAD_TO_LDS),
//   double-buffered in the GEMM so DMA overlaps WMMA, tracked by TENSORcnt.
// ---------------------------------------------------------------------------

typedef __attribute__((ext_vector_type(2))) float v2f;
typedef __attribute__((ext_vector_type(4))) float v4f;
typedef __attribute__((ext_vector_type(8))) float v8f;
typedef __attribute__((ext_vector_type(4))) unsigned int u32x4;
typedef __attribute__((ext_vector_type(4))) int i32x4;
typedef __attribute__((ext_vector_type(8))) int i32x8;

#define WMMA_F32(a, b, c) \
  __builtin_amdgcn_wmma_f32_16x16x4_f32(false, (a), false, (b), (short)0, (c), false, false)

// Problem constants
#define BATCH 8
#define LSEQ 1024
#define SSEQ 1000
#define GDIM 512
#define LDIM 4096
#define HEADS 8
#define DK 64
#define DM 512
#define ODIM 16384
#define MROWS (BATCH * LSEQ)  // 8192

// ---------------------------------------------------------------------------
// TDM descriptor helpers (D# groups per CDNA5 ISA)
// ---------------------------------------------------------------------------
__device__ __forceinline__ unsigned lds_off(const void* p) {
  // LDS aperture: low 32 bits of a generic shared pointer are the
  // wave-relative LDS byte address (ISA §10.2 aperture calculation).
  return (unsigned)(uintptr_t)p;
}

__device__ __forceinline__ u32x4 tdm_group0(unsigned lds_addr, const void* gp) {
  const unsigned long long ga = (unsigned long long)(uintptr_t)gp;
  u32x4 g;
  g.x = 1u;                                   // count=1 (valid), user mode
  g.y = lds_addr;                             // bits 63:32  lds_addr
  g.z = (unsigned)ga;                         // bits 95:64  global_addr[31:0]
  g.w = (unsigned)((ga >> 32) & 0x01ffffffu)  // bits 120:96 global_addr[56:32]
        | (2u << 30);                         // bits 127:126 type=2 ("image")
  return g;
}

// 2D tile: X = tile_d0 elements along stride-1 dim, Y = tile_d1 lines of
// stride0 elements.  tensor_d1 < tile_d1 => rows past the end zero-filled.
// pad_en: insert pad_amount-coded DWORDs into LDS every pad_interval-coded
// DWORDs (intervals: 0=2,1=4,2=8,3=16,4=32,5=64,... amounts: 0=1,1=2,...).
__device__ __forceinline__ i32x8 tdm_group1_2d(
    unsigned tensor_d0, unsigned tensor_d1, unsigned tile_d0, unsigned tile_d1,
    unsigned long long stride0, unsigned pad_interval, unsigned pad_amount,
    bool pad_en) {
  i32x8 g;
  g[0] = (int)((2u << 16)                          // data_size = 2 -> 4 bytes
               | (pad_en ? (1u << 20) : 0u)        // pad_enable
               | (pad_interval << 22)              // bits 24:22
               | (pad_amount << 25));              // bits 31:25
  g[1] = (int)((tensor_d0 & 0xffffu) << 16);       // tensor_dim0[15:0] @ bit48
  g[2] = (int)((tensor_d0 >> 16) | ((tensor_d1 & 0xffffu) << 16));
  g[3] = (int)((tensor_d1 >> 16) | (tile_d0 << 16));  // tile_dim0 @ bit112
  g[4] = (int)(tile_d1 & 0xffffu);                 // tile_dim1; tile_dim2=0
  g[5] = (int)(unsigned)stride0;                   // tensor_dim0_stride[31:0]
  g[6] = (int)((unsigned)(stride0 >> 32) & 0xffffu);
  g[7] = 0;                                        // tensor_dim1_stride = 0
  return g;
}

__device__ __forceinline__ void tdm_load_2d(u32x4 g0, i32x8 g1) {
  const i32x4 z4 = {0, 0, 0, 0};
#if __clang_major__ >= 23
  const i32x8 z8 = {0, 0, 0, 0, 0, 0, 0, 0};
  __builtin_amdgcn_tensor_load_to_lds(g0, g1, z4, z4, z8, 0);
#else
  __builtin_amdgcn_tensor_load_to_lds(g0, g1, z4, z4, 0);
#endif
}

__device__ __forceinline__ float redmax16(float v) {
  v = fmaxf(v, __shfl_xor(v, 1, 32));
  v = fmaxf(v, __shfl_xor(v, 2, 32));
  v = fmaxf(v, __shfl_xor(v, 4, 32));
  v = fmaxf(v, __shfl_xor(v, 8, 32));
  return v;
}
__device__ __forceinline__ float redsum16(float v) {
  v += __shfl_xor(v, 1, 32);
  v += __shfl_xor(v, 2, 32);
  v += __shfl_xor(v, 4, 32);
  v += __shfl_xor(v, 8, 32);
  return v;
}

// ---------------------------------------------------------------------------
// fp32 GEMM + bias:  C[M,N] = A[M,K] @ B[K,N] + bias[N]
// Block tile 64x128, BK=16, 256 threads (8 wave32), wave tile 32x32.
// Tiles DMA'd to LDS by the TDM (wave 0 issues), double-buffered so the
// next tile's DMA overlaps this tile's WMMAs.
// ---------------------------------------------------------------------------
#define AS_STRIDE 18  // 16 + TDM pad(2 DWORDs / 16): 8B-aligned frags, 16 banks
__global__ __launch_bounds__(256) void gemm_bias_f32(
    const float* __restrict__ A, const float* __restrict__ B,
    const float* __restrict__ bias, float* __restrict__ C,
    int M, int N, int K) {
  __shared__ __align__(16) float As[2][64 * AS_STRIDE];
  __shared__ __align__(16) float Bs[2][16 * 128];

  const int tid    = threadIdx.x;
  const int lane   = tid & 31;
  const int wid    = tid >> 5;
  const int lane15 = lane & 15;
  const int hi     = lane >> 4;   // A/B fragment half: K+0,1 vs K+2,3
  const int wm     = wid & 1;     // wave grid 2 (M) x 4 (N)
  const int wn     = wid >> 1;
  const int row0   = blockIdx.x * 64;
  const int col0   = blockIdx.y * 128;
  const int nk     = K >> 4;

  const v8f vz = {0.f, 0.f, 0.f, 0.f, 0.f, 0.f, 0.f, 0.f};
  v8f acc[2][2];
  acc[0][0] = vz; acc[0][1] = vz; acc[1][0] = vz; acc[1][1] = vz;

  // Prologue: DMA tile 0 into buffer 0.
  if (wid == 0) {
    const int k0 = 0;
    tdm_load_2d(tdm_group0(lds_off(&As[0][0]), A + (size_t)row0 * K + k0),
                tdm_group1_2d(16u, (unsigned)(M - row0), 16u, 64u,
                              (unsigned long long)K, 3u, 1u, true));
    tdm_load_2d(tdm_group0(lds_off(&Bs[0][0]), B + (size_t)k0 * N + col0),
                tdm_group1_2d(128u, 16u, 128u, 16u,
                              (unsigned long long)N, 0u, 0u, false));
  }

  for (int i = 0; i < nk; ++i) {
    const int buf = i & 1;
    if (wid == 0) {
      if (i + 1 < nk) {
        const int k1 = (i + 1) << 4;
        tdm_load_2d(tdm_group0(lds_off(&As[buf ^ 1][0]),
                               A + (size_t)row0 * K + k1),
                    tdm_group1_2d(16u, (unsigned)(M - row0), 16u, 64u,
                                  (unsigned long long)K, 3u, 1u, true));
        tdm_load_2d(tdm_group0(lds_off(&Bs[buf ^ 1][0]),
                               B + (size_t)k1 * N + col0),
                    tdm_group1_2d(128u, 16u, 128u, 16u,
                                  (unsigned long long)N, 0u, 0u, false));
        __builtin_amdgcn_s_wait_tensorcnt(2);  // pair i retired, pair i+1 flying
      } else {
        __builtin_amdgcn_s_wait_tensorcnt(0);
      }
    }
    __syncthreads();  // tile i visible to all waves

    const float* as = &As[buf][0];
    const float* bs = &Bs[buf][0];
#pragma unroll
    for (int kk = 0; kk < 4; ++kk) {
      const int kf = kk * 4 + hi * 2;
      v2f a0 = *(const v2f*)&as[(wm * 32 + lane15) * AS_STRIDE + kf];
      v2f a1 = *(const v2f*)&as[(wm * 32 + 16 + lane15) * AS_STRIDE + kf];
#pragma unroll
      for (int nt = 0; nt < 2; ++nt) {
        const int cc = wn * 32 + nt * 16 + lane15;
        v2f bf;
        bf.x = bs[kf * 128 + cc];
        bf.y = bs[(kf + 1) * 128 + cc];
        acc[0][nt] = WMMA_F32(a0, bf, acc[0][nt]);
        acc[1][nt] = WMMA_F32(a1, bf, acc[1][nt]);
      }
    }
    __syncthreads();  // all waves done with buf before TDM overwrites it
  }

  // Epilogue: bias add + store (C layout: VGPR r -> M=r / M=r+8)
#pragma unroll
  for (int mt = 0; mt < 2; ++mt)
#pragma unroll
    for (int nt = 0; nt < 2; ++nt) {
      const int gcol = col0 + wn * 32 + nt * 16 + lane15;
      const float bv = bias[gcol];
#pragma unroll
      for (int r = 0; r < 8; ++r) {
        const int grow = row0 + wm * 32 + mt * 16 + r + hi * 8;
        if (grow < M) C[(size_t)grow * N + gcol] = acc[mt][nt][r] + bv;
      }
    }
}

// ---------------------------------------------------------------------------
// Flash attention per (b, h, 64-row L block).  128 threads (4 wave32),
// each wave owns 16 L-rows (full 64-wide score strip + 16x64 output).
// K/V chunks DMA'd to LDS by the TDM; OOB rows (s >= 1000) zero-filled by
// the tensor_dim1 bound; K chunk gets stride-66 padding from the TDM.
// ---------------------------------------------------------------------------
#define KP_STRIDE 66  // 64 + TDM pad(2 DWORDs / 64)
__global__ __launch_bounds__(128) void flash_attn_f32(
    const float* __restrict__ Qm, const float* __restrict__ Km,
    const float* __restrict__ Vm, float* __restrict__ Om) {
  __shared__ __align__(16) float KP[64 * KP_STRIDE];  // K chunk, then P tile
  __shared__ __align__(16) float Vs[64 * 64];

  const int bh     = blockIdx.x;
  const int b      = bh >> 3;
  const int h      = bh & 7;
  const int l0     = blockIdx.y * 64;
  const int tid    = threadIdx.x;
  const int lane   = tid & 31;
  const int wid    = tid >> 5;  // 0..3 -> rows wid*16..+15
  const int lane15 = lane & 15;
  const int hi     = lane >> 4;

  const float NEG_INF = -__builtin_inff();
  const float scale = 0.125f;  // 1/sqrt(64)

  // Preload Q A-fragments (16) straight into registers.
  v2f qf[16];
  {
    const int qrow = b * LSEQ + l0 + wid * 16 + lane15;
    const float* qp = Qm + (size_t)qrow * DM + h * DK + hi * 2;
#pragma unroll
    for (int kk = 0; kk < 16; ++kk) qf[kk] = *(const v2f*)(qp + kk * 4);
  }

  const v8f vz = {0.f, 0.f, 0.f, 0.f, 0.f, 0.f, 0.f, 0.f};
  v8f o[4];
  o[0] = vz; o[1] = vz; o[2] = vz; o[3] = vz;
  float rm[8], rs[8];
#pragma unroll
  for (int r = 0; r < 8; ++r) { rm[r] = NEG_INF; rs[r] = 0.f; }

  for (int j = 0; j < 16; ++j) {
    const int s0 = j * 64;
    __syncthreads();  // previous chunk fully consumed (KP-as-P and Vs)
    if (wid == 0) {
      const unsigned srem = (unsigned)(SSEQ - s0);  // rows left; rest -> zeros
      tdm_load_2d(tdm_group0(lds_off(&KP[0]), Km + (size_t)s0 * DM + h * DK),
                  tdm_group1_2d(64u, srem, 64u, 64u,
                                (unsigned long long)DM, 5u, 1u, true));
      tdm_load_2d(tdm_group0(lds_off(&Vs[0]), Vm + (size_t)s0 * DM + h * DK),
                  tdm_group1_2d(64u, srem, 64u, 64u,
                                (unsigned long long)DM, 0u, 0u, false));
      __builtin_amdgcn_s_wait_tensorcnt(0);
    }
    __syncthreads();  // chunk visible to all waves

    // Scores: Sc[16 x 64] = Q_tile @ K_chunk^T
    v8f sc[4];
    sc[0] = vz; sc[1] = vz; sc[2] = vz; sc[3] = vz;
#pragma unroll
    for (int kk = 0; kk < 16; ++kk) {
      const int kf = kk * 4 + hi * 2;
      v2f a = qf[kk];
#pragma unroll
      for (int nt = 0; nt < 4; ++nt) {
        const int coln = nt * 16 + lane15;  // s within chunk
        v2f bf = *(const v2f*)&KP[coln * KP_STRIDE + kf];
        sc[nt] = WMMA_F32(a, bf, sc[nt]);
      }
    }

    // Scale + mask invalid columns.
#pragma unroll
    for (int nt = 0; nt < 4; ++nt) {
      const bool ok = (s0 + nt * 16 + lane15) < SSEQ;
#pragma unroll
      for (int r = 0; r < 8; ++r)
        sc[nt][r] = ok ? sc[nt][r] * scale : NEG_INF;
    }

    // Online softmax (xor masks <16 keep half-waves separate, matching the
    // C layout: VGPR r holds row r in lanes 0-15 and row r+8 in lanes 16-31).
#pragma unroll
    for (int r = 0; r < 8; ++r) {
      float cm = fmaxf(fmaxf(sc[0][r], sc[1][r]), fmaxf(sc[2][r], sc[3][r]));
      cm = redmax16(cm);
      const float nm = fmaxf(rm[r], cm);
      const float alpha = __expf(rm[r] - nm);
      float p0 = __expf(sc[0][r] - nm);
      float p1 = __expf(sc[1][r] - nm);
      float p2 = __expf(sc[2][r] - nm);
      float p3 = __expf(sc[3][r] - nm);
      sc[0][r] = p0; sc[1][r] = p1; sc[2][r] = p2; sc[3][r] = p3;
      const float csum = redsum16(p0 + p1 + p2 + p3);
      rs[r] = rs[r] * alpha + csum;
      rm[r] = nm;
      o[0][r] *= alpha; o[1][r] *= alpha; o[2][r] *= alpha; o[3][r] *= alpha;
    }

    __syncthreads();  // all waves done reading KP as K chunk
    // Stage P (C-layout -> LDS); a wave writes only its own 16 rows.
#pragma unroll
    for (int nt = 0; nt < 4; ++nt)
#pragma unroll
      for (int r = 0; r < 8; ++r)
        KP[(wid * 16 + r + hi * 8) * KP_STRIDE + nt * 16 + lane15] = sc[nt][r];

    // O += P @ V  (A-frag from the wave's own P rows, B-frag from Vs).
    const int arow = wid * 16 + lane15;
#pragma unroll
    for (int kk = 0; kk < 16; ++kk) {
      const int kf = kk * 4 + hi * 2;
      v2f a = *(const v2f*)&KP[arow * KP_STRIDE + kf];
#pragma unroll
      for (int nt = 0; nt < 4; ++nt) {
        const int colv = nt * 16 + lane15;
        v2f bf;
        bf.x = Vs[kf * 64 + colv];
        bf.y = Vs[(kf + 1) * 64 + colv];
        o[nt] = WMMA_F32(a, bf, o[nt]);
      }
    }
  }

  // Normalize + store: out[b, l, h*64+e]
#pragma unroll
  for (int nt = 0; nt < 4; ++nt) {
    const int e = nt * 16 + lane15;
#pragma unroll
    for (int r = 0; r < 8; ++r) {
      const int row = wid * 16 + r + hi * 8;
      const size_t grow = (size_t)(b * LSEQ + l0 + row);
      Om[grow * DM + h * DK + e] = o[nt][r] / rs[r];
    }
  }
}

// ---------------------------------------------------------------------------
// Launch: QKV projections -> flash attention -> output projection
// ---------------------------------------------------------------------------
extern "C" void kernel_launch(void* const* d_in, const int* in_sizes, int n_in,
                              void* d_out, int out_size, void* d_ws, size_t ws_size,
                              hipStream_t stream) {
  const float* tgt = (const float*)d_in[0];  // [8, 1024, 512]
  const float* src = (const float*)d_in[1];  // [1000, 4096]
  const float* val = (const float*)d_in[2];  // [1000, 4096]
  const float* Wq  = (const float*)d_in[3];
  const float* bq  = (const float*)d_in[4];
  const float* Wk  = (const float*)d_in[5];
  const float* bk  = (const float*)d_in[6];
  const float* Wv  = (const float*)d_in[7];
  const float* bv  = (const float*)d_in[8];
  const float* Wo  = (const float*)d_in[9];
  const float* bo  = (const float*)d_in[10];
  float* out = (float*)d_out;                // [8192, 16384]

  float* Q  = (float*)d_ws;                  // [8192, 512]
  float* Kp = Q  + (size_t)MROWS * DM;       // [1000, 512]
  float* Vp = Kp + (size_t)SSEQ * DM;        // [1000, 512]
  float* AO = Vp + (size_t)SSEQ * DM;        // [8192, 512]

  // Q projection: [8192,512] @ [512,512]
  gemm_bias_f32<<<dim3(MROWS / 64, DM / 128), 256, 0, stream>>>(
      tgt, Wq, bq, Q, MROWS, DM, GDIM);
  // K projection: [1000,4096] @ [4096,512]
  gemm_bias_f32<<<dim3((SSEQ + 63) / 64, DM / 128), 256, 0, stream>>>(
      src, Wk, bk, Kp, SSEQ, DM, LDIM);
  // V projection
  gemm_bias_f32<<<dim3((SSEQ + 63) / 64, DM / 128), 256, 0, stream>>>(
      val, Wv, bv, Vp, SSEQ, DM, LDIM);
  // Attention: grid (B*H, L/64)
  flash_attn_f32<<<dim3(BATCH * HEADS, LSEQ / 64), 128, 0, stream>>>(Q, Kp, Vp, AO);
  // Output projection: [8192,512] @ [512,16384]
  gemm_bias_f32<<<dim3(MROWS / 64, ODIM / 128), 256, 0, stream>>>(
      AO, Wo, bo, out, MROWS, ODIM, DM);
}